// LearnableLPDistance_85710367359025
// MI455X (gfx1250) — compile-verified
//
#include <hip/hip_runtime.h>
#include <hip/hip_bf16.h>
#include <stdint.h>

// Problem shape (fixed by reference): A,B = [1, 8, 512, 64] f32, out = [1, 8, 512, 512] f32
#define HEADS 8
#define NN    512
#define DD    64
#define TILE  64
#define LSTR  68   // padded LDS row stride (floats); 68*4 = 272 B keeps 16B alignment

#define EPS 1e-10f

// |x|^e for x >= 0 via hardware log2/exp2 (v_log_f32 is log2, v_exp_f32 is exp2).
// x == 0 -> log2 = -inf -> exp2 = 0, which matches pow(0, e) for e > 0.
__device__ __forceinline__ float fast_pow_pos(float x, float e) {
    return __builtin_amdgcn_exp2f(e * __builtin_amdgcn_logf(x));
}

__device__ __forceinline__ float softplus_p(const float* __restrict__ p_raw) {
    float x = p_raw[0];
    float p = (x > 20.0f) ? x : log1pf(expf(x));
    return p + EPS;
}

// ---------------------------------------------------------------------------
// Pass 1: Lp-normalize all rows of A and B into workspace.
// ws layout: rows 0..4095 = An (h*512+n), rows 4096..8191 = Bn.
// One wave (32 lanes) per row; each lane holds 2 consecutive floats (float2).
// ---------------------------------------------------------------------------
__global__ __launch_bounds__(256) void lp_normalize_kernel(
    const float* __restrict__ A, const float* __restrict__ B,
    const float* __restrict__ p_raw, float* __restrict__ ws) {
    const float p     = softplus_p(p_raw);
    const float inv_p = 1.0f / p;

    const int wave = threadIdx.x >> 5;
    const int lane = threadIdx.x & 31;
    const int row  = blockIdx.x * 8 + wave;           // 0 .. 8191

    const float* src = (row < HEADS * NN) ? (A + (size_t)row * DD)
                                          : (B + (size_t)(row - HEADS * NN) * DD);
    float* dst = ws + (size_t)row * DD;

    float2 v = ((const float2*)src)[lane];
    float  s = fast_pow_pos(fabsf(v.x), p) + fast_pow_pos(fabsf(v.y), p);

    // wave32 butterfly reduction
    #pragma unroll
    for (int off = 16; off > 0; off >>= 1)
        s += __shfl_xor(s, off, 32);

    const float norm = fast_pow_pos(s, inv_p) + EPS;
    const float r    = 1.0f / norm;
    float2 o; o.x = v.x * r; o.y = v.y * r;
    ((float2*)dst)[lane] = o;
}

// ---------------------------------------------------------------------------
// Pass 2: pairwise Lp distance over 64x64 output tiles.
// Block = 256 threads; thread (tn, tm) in a 16x16 grid owns a 4x4 register
// sub-tile. Tiles are staged into LDS with gfx1250 async global->LDS copies.
// ---------------------------------------------------------------------------
__global__ __launch_bounds__(256) void lp_distance_kernel(
    const float* __restrict__ ws, const float* __restrict__ p_raw,
    float* __restrict__ out) {
    __shared__ float At[TILE * LSTR];
    __shared__ float Bt[TILE * LSTR];

    const float p     = softplus_p(p_raw);
    const float inv_p = 1.0f / p;

    const int tid = threadIdx.x;
    const int tm  = tid & 15;
    const int tn  = tid >> 4;
    const int h   = blockIdx.z;
    const int n0  = blockIdx.y * TILE;
    const int m0  = blockIdx.x * TILE;

    const float* Abase = ws + ((size_t)h * NN + n0) * DD;
    const float* Bbase = ws + ((size_t)(HEADS * NN) + (size_t)h * NN + m0) * DD;

    // Async-copy both 64x64 f32 tiles (16 KB each, row-contiguous in global)
    // into padded LDS rows. 4 x b128 per thread per tile.
    #pragma unroll
    for (int k = 0; k < 4; ++k) {
        const int i   = k * 256 + tid;
        const int row = i >> 4;          // 0..63
        const int c   = i & 15;          // 16B chunk within the 256B row
        uint32_t aoff = (uint32_t)(uintptr_t)(void*)&At[row * LSTR + c * 4];
        uint32_t boff = (uint32_t)(uintptr_t)(void*)&Bt[row * LSTR + c * 4];
        const float* ga = Abase + row * DD + c * 4;
        const float* gb = Bbase + row * DD + c * 4;
        asm volatile("global_load_async_to_lds_b128 %0, %1, off"
                     :: "v"(aoff), "v"(ga) : "memory");
        asm volatile("global_load_async_to_lds_b128 %0, %1, off"
                     :: "v"(boff), "v"(gb) : "memory");
    }
    asm volatile("s_wait_asynccnt 0" ::: "memory");
    __syncthreads();

    float acc[4][4];
    #pragma unroll
    for (int i = 0; i < 4; ++i)
        #pragma unroll
        for (int j = 0; j < 4; ++j)
            acc[i][j] = 0.0f;

    const float* Ar = &At[(tn * 4) * LSTR];
    const float* Br = &Bt[(tm * 4) * LSTR];

    // Trans-bound inner loop: 8 LDS reads feed 16 |a-b|^p updates (32 trans).
    for (int d = 0; d < DD; ++d) {
        float a[4], b[4];
        #pragma unroll
        for (int i = 0; i < 4; ++i) a[i] = Ar[i * LSTR + d];
        #pragma unroll
        for (int j = 0; j < 4; ++j) b[j] = Br[j * LSTR + d];
        #pragma unroll
        for (int i = 0; i < 4; ++i)
            #pragma unroll
            for (int j = 0; j < 4; ++j) {
                float t = fabsf(a[i] - b[j]);
                acc[i][j] += fast_pow_pos(t, p);
            }
    }

    // Epilogue: dist = acc^(1/p); out = 1 - dist/2. Coalesced float4 stores.
    #pragma unroll
    for (int i = 0; i < 4; ++i) {
        float4 o;
        o.x = 1.0f - 0.5f * fast_pow_pos(acc[i][0], inv_p);
        o.y = 1.0f - 0.5f * fast_pow_pos(acc[i][1], inv_p);
        o.z = 1.0f - 0.5f * fast_pow_pos(acc[i][2], inv_p);
        o.w = 1.0f - 0.5f * fast_pow_pos(acc[i][3], inv_p);
        const int n = n0 + tn * 4 + i;
        float* orow = out + ((size_t)h * NN + n) * NN + (m0 + tm * 4);
        *(float4*)orow = o;
    }
}

extern "C" void kernel_launch(void* const* d_in, const int* in_sizes, int n_in,
                              void* d_out, int out_size, void* d_ws, size_t ws_size,
                              hipStream_t stream) {
    const float* A     = (const float*)d_in[0];
    const float* B     = (const float*)d_in[1];
    const float* p_raw = (const float*)d_in[2];
    float*       out   = (float*)d_out;
    float*       ws    = (float*)d_ws;   // needs 2 * 8 * 512 * 64 * 4 B = 2 MB

    // Pass 1: 8192 rows, 8 rows (waves) per 256-thread block.
    lp_normalize_kernel<<<dim3(1024), dim3(256), 0, stream>>>(A, B, p_raw, ws);

    // Pass 2: 8x8 tiles of 64x64 per head, 8 heads.
    lp_distance_kernel<<<dim3(8, 8, 8), dim3(256), 0, stream>>>(ws, p_raw, out);
}